// WhdrTestLoss_Paper_15994458211240
// MI455X (gfx1250) — compile-verified
//
#include <hip/hip_runtime.h>
#include <math.h>

#define DELTA_F 0.1f
#define EPS_F   1e-10f

typedef __attribute__((ext_vector_type(2))) float v2f;
typedef __attribute__((ext_vector_type(8))) float v8f;
typedef __attribute__((ext_vector_type(2))) int   v2i;   // clang ext-vector: OK for nontemporal builtin

// Exact f32 wave(32)-wide sum using V_WMMA_F32_16X16X4_F32.
// A: lane's value in first A VGPR, 0 in second -> every A element holds each
// lane's value exactly once (plus zeros). B: all ones. D[m][n] = rowsum(m).
// Each lane sums its 8 D VGPRs (its column, half the rows), then one
// shfl_xor(16) combines the two half-wave row groups -> all lanes hold the
// exact f32 sum of all 32 lanes' inputs. Layout-robust: with B==1 everywhere
// and the unused A VGPR zeroed, any bijective element->(m,k) striping gives
// the same total.
__device__ __forceinline__ float wave_sum_wmma(float val) {
  v2f a; a.x = val;  a.y = 0.0f;
  v2f b; b.x = 1.0f; b.y = 1.0f;
  v8f c = {};
  c = __builtin_amdgcn_wmma_f32_16x16x4_f32(
      /*neg_a=*/false, a, /*neg_b=*/false, b,
      /*c_mod=*/(short)0, c, /*reuse_a=*/false, /*reuse_b=*/false);
  float s = c[0] + c[1] + c[2] + c[3] + c[4] + c[5] + c[6] + c[7];
  s += __shfl_xor(s, 16, 32);
  return s;
}

// Pass 1: stream 3xHxW input once, write HW-float grayscale table.
// float4-vectorized (HW is a multiple of 4). Table stays L2-resident (16 MB
// vs 192 MB L2), so the 2M random gathers in pass 2 never touch HBM.
__global__ __launch_bounds__(256) void precompute_v_kernel(
    const float4* __restrict__ in, float4* __restrict__ V, int hw4) {
  int stride = gridDim.x * blockDim.x;
  for (int i = blockIdx.x * blockDim.x + threadIdx.x; i < hw4; i += stride) {
    float4 a = in[i];
    float4 b = in[i + hw4];
    float4 c = in[i + 2 * hw4];
    float4 o;
    o.x = ((a.x + b.x + c.x) / 3.0f * 255.0f + 1.0f) / 256.0f;
    o.y = ((a.y + b.y + c.y) / 3.0f * 255.0f + 1.0f) / 256.0f;
    o.z = ((a.z + b.z + c.z) / 3.0f * 255.0f + 1.0f) / 256.0f;
    o.w = ((a.w + b.w + c.w) / 3.0f * 255.0f + 1.0f) / 256.0f;
    V[i] = o;
  }
}

// Pass 2: gather + classify + per-block deterministic partial sums.
// PRE=true: img = precomputed HW grayscale table (preferred).
// PRE=false: img = raw 3xHxW input, grayscale computed per gather (fallback
// when workspace can't hold the table).
template <bool PRE>
__global__ __launch_bounds__(256) void whdr_main_kernel(
    const float* __restrict__ img,
    const v2i*   __restrict__ xy1,
    const v2i*   __restrict__ xy2,
    const int*   __restrict__ darker,
    const float* __restrict__ weight,
    float2* __restrict__ partials,
    int N, int W, int HW) {
  float pl = 0.0f, pw = 0.0f;
  int stride = gridDim.x * blockDim.x;
  for (int i = blockIdx.x * blockDim.x + threadIdx.x; i < N; i += stride) {
    // Once-read streams: non-temporal so they don't evict the gather table.
    v2i   p1 = __builtin_nontemporal_load(&xy1[i]);
    v2i   p2 = __builtin_nontemporal_load(&xy2[i]);
    int   d  = __builtin_nontemporal_load(&darker[i]);
    float w  = __builtin_nontemporal_load(&weight[i]);
    int i1 = p1.y * W + p1.x;   // xy[:,1] is row, xy[:,0] is col
    int i2 = p2.y * W + p2.x;
    float r1, r2;
    if (PRE) {
      r1 = img[i1];
      r2 = img[i2];
    } else {
      float s1 = img[i1] + img[i1 + HW] + img[i1 + 2 * HW];
      float s2 = img[i2] + img[i2 + HW] + img[i2 + 2 * HW];
      r1 = (s1 / 3.0f * 255.0f + 1.0f) / 256.0f;
      r2 = (s2 / 3.0f * 255.0f + 1.0f) / 256.0f;
    }
    int alg = (r2 / (r1 + EPS_F) > 1.0f + DELTA_F)
                  ? 1
                  : ((r1 / (r2 + EPS_F) > 1.0f + DELTA_F) ? 2 : 0);
    pw += w;
    pl += (alg != d) ? w : 0.0f;
  }

  // Wave-level exact f32 sum via WMMA (EXEC is all-ones here: loop has
  // reconverged and the block is fully populated).
  float wl = wave_sum_wmma(pl);
  float ww = wave_sum_wmma(pw);

  __shared__ float sL[8], sW[8];
  int lane = threadIdx.x & 31;
  int wave = threadIdx.x >> 5;
  if (lane == 0) { sL[wave] = wl; sW[wave] = ww; }
  __syncthreads();
  if (threadIdx.x == 0) {
    float tl = 0.0f, tw = 0.0f;
#pragma unroll
    for (int j = 0; j < 8; ++j) { tl += sL[j]; tw += sW[j]; }
    partials[blockIdx.x] = make_float2(tl, tw);  // no atomics: deterministic
  }
}

// Pass 3: single block reduces all block partials (fixed assignment ->
// deterministic), WMMA wave reduction again, then writes loss/total_weight.
__global__ __launch_bounds__(256) void whdr_finalize_kernel(
    const float2* __restrict__ partials, int nb, float* __restrict__ out) {
  float tl = 0.0f, tw = 0.0f;
  for (int i = threadIdx.x; i < nb; i += 256) {
    float2 p = partials[i];
    tl += p.x;
    tw += p.y;
  }
  float wl = wave_sum_wmma(tl);
  float ww = wave_sum_wmma(tw);
  __shared__ float sL[8], sW[8];
  int lane = threadIdx.x & 31;
  int wave = threadIdx.x >> 5;
  if (lane == 0) { sL[wave] = wl; sW[wave] = ww; }
  __syncthreads();
  if (threadIdx.x == 0) {
    float TL = 0.0f, TW = 0.0f;
#pragma unroll
    for (int j = 0; j < 8; ++j) { TL += sL[j]; TW += sW[j]; }
    out[0] = TL / TW;
  }
}

extern "C" void kernel_launch(void* const* d_in, const int* in_sizes, int n_in,
                              void* d_out, int out_size, void* d_ws, size_t ws_size,
                              hipStream_t stream) {
  const float* v_input = (const float*)d_in[0];
  const v2i*   xy1     = (const v2i*)d_in[1];
  const v2i*   xy2     = (const v2i*)d_in[2];
  const int*   darker  = (const int*)d_in[3];
  const float* weight  = (const float*)d_in[4];
  float* out = (float*)d_out;

  const int  N  = in_sizes[3];                 // darker has N entries
  const long hw = (long)in_sizes[0] / 3;       // H*W
  const int  W  = (int)(sqrt((double)hw) + 0.5);
  const int  HW = (int)hw;

  // Workspace layout: [0, 32KB) block partials (float2), [32KB, ...) V table.
  const size_t part_off = 0;
  const size_t v_off    = 32768;
  const size_t v_bytes  = (size_t)hw * sizeof(float);

  int nb = (N + 255) / 256;
  if (nb > 4096) nb = 4096;
  // Partials must fit in workspace (4096 blocks -> 32 KB).
  size_t max_nb = ws_size / sizeof(float2);
  if ((size_t)nb > max_nb) nb = (int)max_nb;
  if (nb < 1) nb = 1;

  float2* partials = (float2*)((char*)d_ws + part_off);
  const bool use_table = (ws_size >= v_off + v_bytes) && (nb <= 4096);

  if (use_table) {
    float* V = (float*)((char*)d_ws + v_off);
    int hw4 = HW / 4;
    int pb  = (hw4 + 255) / 256;
    if (pb > 4096) pb = 4096;
    precompute_v_kernel<<<pb, 256, 0, stream>>>(
        (const float4*)v_input, (float4*)V, hw4);
    whdr_main_kernel<true><<<nb, 256, 0, stream>>>(
        V, xy1, xy2, darker, weight, partials, N, W, HW);
  } else {
    whdr_main_kernel<false><<<nb, 256, 0, stream>>>(
        v_input, xy1, xy2, darker, weight, partials, N, W, HW);
  }
  whdr_finalize_kernel<<<1, 256, 0, stream>>>(partials, nb, out);
}